// QSANNAdapter_14456859918886
// MI455X (gfx1250) — compile-verified
//
#include <hip/hip_runtime.h>
#include <hip/hip_bf16.h>
#include <math.h>

// ---------------------------------------------------------------------------
// QSANN distance-kernel attention for gfx1250 (MI455X), flash-style streaming.
// Stage 1: q/k projection (8->4) + LayerNorm + row squared-norms -> workspace.
// Stage 2: per-wave 16-row tile; stream j in blocks of 16 using
//          V_WMMA_F32_16X16X4_F32 for both k*q^T (scores, transposed) and
//          V^T*P^T (output, transposed -> matches [B_H, C_HEAD, S] layout).
//          C-layout -> B-operand relayout of P^T done with ds_bpermute shuffles.
//          exp(-dist/tau) via raw v_exp_f32 (argument provably in [-18, ~0]).
// ---------------------------------------------------------------------------

typedef float v2f __attribute__((ext_vector_type(2)));
typedef float v4f __attribute__((ext_vector_type(4)));
typedef float v8f __attribute__((ext_vector_type(8)));

static constexpr int NQ    = 8;      // n_qubits
static constexpr int QKD   = 4;      // qk_dim
static constexpr int BH    = 16;     // B_H
static constexpr int SEQ   = 2048;   // S
static constexpr int DH    = 64;     // C_head
static constexpr int NROWS = BH * SEQ;  // 32768

// Raw hardware exp2: argument here is bounded in [-18, ~0] (post-LayerNorm
// norms <= ~4, tau ~= 1.31), far from the -126 denorm threshold, so the
// libm clamp/ldexp fixup sequence is dead weight.
static __device__ __forceinline__ float fast_exp2(float x) {
#if __has_builtin(__builtin_amdgcn_exp2f)
  return __builtin_amdgcn_exp2f(x);
#else
  float r;
  asm volatile("v_exp_f32 %0, %1\n\tv_nop" : "=v"(r) : "v"(x));
  return r;
#endif
}

// ---------------- Stage 1: projection + LayerNorm + squared norms -----------
__global__ __launch_bounds__(256) void qsann_proj_ln(
    const float* __restrict__ q_all, const float* __restrict__ k_all,
    const float* __restrict__ qw, const float* __restrict__ qbias,
    const float* __restrict__ kw, const float* __restrict__ kbias,
    const float* __restrict__ lnw, const float* __restrict__ lnb,
    float* __restrict__ fq, float* __restrict__ nq,
    float* __restrict__ fk, float* __restrict__ nk)
{
  int r = blockIdx.x * blockDim.x + threadIdx.x;   // 0 .. 2*NROWS-1
  bool isK = r >= NROWS;
  int row  = isK ? (r - NROWS) : r;
  const float* __restrict__ x  = (isK ? k_all : q_all) + (size_t)row * NQ;
  const float* __restrict__ W  = isK ? kw : qw;
  const float* __restrict__ Bv = isK ? kbias : qbias;

  float y[QKD];
#pragma unroll
  for (int d = 0; d < QKD; ++d) {
    float acc = Bv[d];
#pragma unroll
    for (int i = 0; i < NQ; ++i) acc = fmaf(x[i], W[d * NQ + i], acc);
    y[d] = acc;
  }
  float mu = 0.25f * (y[0] + y[1] + y[2] + y[3]);
  float var = 0.0f;
#pragma unroll
  for (int d = 0; d < QKD; ++d) { float dv = y[d] - mu; var = fmaf(dv, dv, var); }
  var *= 0.25f;
  float rs = rsqrtf(var + 1e-5f);

  v4f o;
  float n = 0.0f;
#pragma unroll
  for (int d = 0; d < QKD; ++d) {
    float v = fmaf((y[d] - mu) * rs, lnw[d], lnb[d]);
    o[d] = v;
    n = fmaf(v, v, n);
  }
  *(v4f*)((isK ? fk : fq) + (size_t)row * QKD) = o;
  (isK ? nk : nq)[row] = n;
}

// ---------------- Stage 2: streaming distance-attention ---------------------
__global__ __launch_bounds__(256) void qsann_attn(
    const float* __restrict__ fq, const float* __restrict__ nq,
    const float* __restrict__ fk, const float* __restrict__ nk,
    const float* __restrict__ xh, const float* __restrict__ raw_tau,
    float* __restrict__ out)
{
  const int wave = threadIdx.x >> 5;
  const int lane = threadIdx.x & 31;
  const int l16  = lane & 15;
  const bool lo  = lane < 16;
  const int hi8  = (lane >> 4) << 3;     // 0 for lanes 0-15, 8 for lanes 16-31
  const int doff = lo ? 0 : 2;           // K-pair offset per A/B f32 layout

  const int b     = blockIdx.x >> 4;
  const int stile = blockIdx.x & 15;
  const int s0    = (stile * 8 + wave) * 16;   // this wave's 16 s-rows

  // tau_eff = softplus(raw_tau) + 1e-9 ; fold log2(e) into the exp2 scale
  float rt    = raw_tau[0];
  float sp    = (rt > 20.0f) ? rt : log1pf(__expf(rt));
  float scale = 1.4426950408889634f / (sp + 1e-9f);

  // B-operand (q^T, 4x16): lane holds q[s0+l16][doff], q[s0+l16][doff+1]
  const int srow = s0 + l16;
  v2f qb = *(const v2f*)(fq + (size_t)srow * QKD + doff);
  const float qnv = nq[srow];

  const float* __restrict__ xb = xh + (size_t)b * DH * SEQ;

  v8f acc[4];                    // out^T accumulators: 4 d-tiles of 16
  acc[0] = {}; acc[1] = {}; acc[2] = {}; acc[3] = {};
  float rsum = 0.0f;             // partial alpha row-sum for s = l16, half hi8

  for (int jb = 0; jb < SEQ; jb += 16) {
    if (jb + 16 < SEQ)
      __builtin_prefetch((const void*)(fk + (size_t)(jb + 16 + l16) * QKD), 0, 1);

    // A-operand (k tile, 16x4): lane holds k[jb+l16][doff], k[jb+l16][doff+1]
    v2f ka = *(const v2f*)(fk + (size_t)(jb + l16) * QKD + doff);
    // k squared-norms for this lane's 8 score rows (j = jb + hi8 + r)
    v4f kn0 = *(const v4f*)(nk + jb + hi8);
    v4f kn1 = *(const v4f*)(nk + jb + hi8 + 4);

    // Scores transposed: S^T[j][s] = k_j . q_s  (M=j, K=d(4), N=s)
    v8f sc = {};
    sc = __builtin_amdgcn_wmma_f32_16x16x4_f32(false, ka, false, qb,
                                               (short)0, sc, false, false);

    // P^T[j][s] = exp(-(|q_s|^2 + |k_j|^2 - 2 q.k) / tau), in C-layout
    float w[8];
#pragma unroll
    for (int r = 0; r < 8; ++r) {
      float knv = (r < 4) ? kn0[r] : kn1[r - 4];
      w[r] = fast_exp2((2.0f * sc[r] - qnv - knv) * scale);
      rsum += w[r];
    }

    // out^T += V^T * P^T : K(j)=16 split into 4 chunks of 4; 4 d-tiles of 16
#pragma unroll
    for (int kc = 0; kc < 4; ++kc) {
      const int base = (kc & 1) * 4;              // source vgpr group in w[]
      const int src  = l16 + ((kc & 2) ? 16 : 0); // source lane half
      float t0 = __shfl(w[base + 0], src, 32);
      float t1 = __shfl(w[base + 1], src, 32);
      float t2 = __shfl(w[base + 2], src, 32);
      float t3 = __shfl(w[base + 3], src, 32);
      v2f pb;                                     // B-operand P^T chunk (4x16)
      pb[0] = lo ? t0 : t2;
      pb[1] = lo ? t1 : t3;

      // A-operand V^T chunks (16x4): xh[b][d][j], contiguous float2 in j
      const float* vp = xb + (size_t)l16 * SEQ + (jb + kc * 4 + doff);
      v2f va0 = *(const v2f*)(vp);
      v2f va1 = *(const v2f*)(vp + 16 * SEQ);
      v2f va2 = *(const v2f*)(vp + 32 * SEQ);
      v2f va3 = *(const v2f*)(vp + 48 * SEQ);
      acc[0] = __builtin_amdgcn_wmma_f32_16x16x4_f32(false, va0, false, pb,
                                                     (short)0, acc[0], false, false);
      acc[1] = __builtin_amdgcn_wmma_f32_16x16x4_f32(false, va1, false, pb,
                                                     (short)0, acc[1], false, false);
      acc[2] = __builtin_amdgcn_wmma_f32_16x16x4_f32(false, va2, false, pb,
                                                     (short)0, acc[2], false, false);
      acc[3] = __builtin_amdgcn_wmma_f32_16x16x4_f32(false, va3, false, pb,
                                                     (short)0, acc[3], false, false);
    }
  }

  // total alpha row-sum for s = l16: combine the two lane halves
  float tot = rsum + __shfl_xor(rsum, 16, 32);
  float inv = 1.0f / (tot + 1e-9f);

  // out^T tiles are already in [d][s] C-layout == reference output layout
  float* __restrict__ ob = out + (size_t)b * DH * SEQ;
#pragma unroll
  for (int nt = 0; nt < 4; ++nt) {
#pragma unroll
    for (int r = 0; r < 8; ++r) {
      int d = nt * 16 + hi8 + r;
      ob[(size_t)d * SEQ + s0 + l16] = acc[nt][r] * inv;
    }
  }
}

// ---------------------------------------------------------------------------
extern "C" void kernel_launch(void* const* d_in, const int* in_sizes, int n_in,
                              void* d_out, int out_size, void* d_ws, size_t ws_size,
                              hipStream_t stream) {
  const float* q_all = (const float*)d_in[0];
  const float* k_all = (const float*)d_in[1];
  const float* xh    = (const float*)d_in[2];
  const float* qw    = (const float*)d_in[3];
  const float* qb    = (const float*)d_in[4];
  const float* kw    = (const float*)d_in[5];
  const float* kb    = (const float*)d_in[6];
  const float* lnw   = (const float*)d_in[7];
  const float* lnb   = (const float*)d_in[8];
  const float* tau   = (const float*)d_in[9];

  // Workspace layout (floats): fq[32768*4] nq[32768] fk[32768*4] nk[32768]
  float* ws  = (float*)d_ws;
  float* fq  = ws;
  float* nqv = fq  + (size_t)NROWS * QKD;
  float* fk  = nqv + NROWS;
  float* nkv = fk  + (size_t)NROWS * QKD;

  qsann_proj_ln<<<(2 * NROWS) / 256, 256, 0, stream>>>(
      q_all, k_all, qw, qb, kw, kb, lnw, lnb, fq, nqv, fk, nkv);

  qsann_attn<<<BH * (SEQ / 128), 256, 0, stream>>>(
      fq, nqv, fk, nkv, xh, tau, (float*)d_out);
}